// TaskCRF_24790551232618
// MI455X (gfx1250) — compile-verified
//
#include <hip/hip_runtime.h>
#include <hip/hip_bf16.h>
#include <math.h>

// B=64, S=512, D=1024, L=9
#define BB 64
#define SS 512
#define DD 1024
#define LL 9
#define LPAD 16   // N-tile width for WMMA (cols 9..15 are zero)

typedef __attribute__((ext_vector_type(2))) float v2f;
typedef __attribute__((ext_vector_type(8))) float v8f;

// ---------------------------------------------------------------------------
// Kernel 1: logits = inputs @ W + b  via V_WMMA_F32_16X16X4_F32
//   grid: 256 blocks x 256 threads (8 waves); each wave: one 16-row M tile,
//   K-loop over D=1024 in steps of 4.  W staged (zero-padded to 16 cols) in LDS.
// ---------------------------------------------------------------------------
__global__ __launch_bounds__(256)
void crf_gemm_logits_kernel(const float* __restrict__ A,   // (B*S, D)
                            const float* __restrict__ W,   // (D, L)
                            const float* __restrict__ bias,// (L,)
                            float* __restrict__ out)       // out[0]=nll, out+1 = logits (B*S, L)
{
    __shared__ float sW[DD * LPAD];          // 64 KB, zero-padded W

    const int tid = threadIdx.x;

    // cooperative zero-padded staging of W into LDS (no divergence: ternary)
    for (int idx = tid; idx < DD * LPAD; idx += 256) {
        const int k = idx >> 4;
        const int n = idx & 15;
        sW[idx] = (n < LL) ? W[k * LL + n] : 0.0f;
    }

    if (blockIdx.x == 0 && tid == 0) out[0] = 0.0f;   // init NLL accumulator

    __syncthreads();

    const int wave = tid >> 5;
    const int lane = tid & 31;
    const int n    = lane & 15;                 // N index (B/C/D fragment column)
    const int koff = (lane >> 4) << 1;          // lanes 16-31 hold K = {2,3}

    const int tileRow = blockIdx.x * 128 + wave * 16;   // 256*128 = 32768 rows exactly
    const int rowA    = tileRow + (lane & 15);          // A fragment row (M = lane%16)
    const float* __restrict__ arow = A + (size_t)rowA * DD;

    v8f acc = {0.f, 0.f, 0.f, 0.f, 0.f, 0.f, 0.f, 0.f};

#pragma unroll 4
    for (int k = 0; k < DD; k += 4) {
        // A 16x4 fragment: lane holds A[row][k+koff], A[row][k+koff+1]  (8B load)
        const v2f a = *(const v2f*)(arow + k + koff);
        // B 4x16 fragment (mirrored layout): lane holds W[k+koff][n], W[k+koff+1][n]
        v2f b;
        b.x = sW[(k + koff) * LPAD + n];
        b.y = sW[(k + koff + 1) * LPAD + n];
        acc = __builtin_amdgcn_wmma_f32_16x16x4_f32(
                  /*neg_a=*/false, a, /*neg_b=*/false, b,
                  /*c_mod=*/(short)0, acc, /*reuse_a=*/false, /*reuse_b=*/false);
    }

    // epilogue: C/D layout: VGPR v -> M = v (lanes 0-15) or v+8 (lanes 16-31), N = lane%16
    if (n < LL) {
        const float bn = bias[n];
        const int moff = (lane >> 4) << 3;
        float* __restrict__ logits = out + 1;
#pragma unroll
        for (int v = 0; v < 8; ++v) {
            const int m = v + moff;
            logits[(size_t)(tileRow + m) * LL + n] = acc[v] + bn;
        }
    }
}

// ---------------------------------------------------------------------------
// Kernel 2: CRF NLL.  One wave per batch element (64 blocks x 32 threads).
//   numerator: parallel gather over s, warp reduction.
//   alpha scan: lane j<9 holds alpha[j]; cross-lane broadcast via __shfl.
// ---------------------------------------------------------------------------
__global__ __launch_bounds__(32)
void crf_forward_kernel(const int* __restrict__ targets,      // (B,S)
                        const int* __restrict__ seq_lens,     // (B,)
                        const float* __restrict__ T,          // (L,L)
                        float* __restrict__ out)              // out[0]=nll, out+1=logits
{
    const int b    = blockIdx.x;
    const int lane = threadIdx.x;
    const int seqlen = seq_lens[b];

    const float* __restrict__ lg  = out + 1 + (size_t)b * SS * LL;
    const int*   __restrict__ tgt = targets + (size_t)b * SS;

    // ---- numerator: emissions + transitions along the gold path ----
    float num = 0.0f;
    for (int s = lane; s < SS; s += 32) {
        if (s < seqlen) {
            const int ts = tgt[s];
            num += lg[s * LL + ts];
            if (s >= 1) num += T[tgt[s - 1] * LL + ts];
        }
    }
#pragma unroll
    for (int off = 16; off > 0; off >>= 1) num += __shfl_xor(num, off);

    // ---- alpha recurrence (forward algorithm) ----
    const int jj = (lane < LL) ? lane : (LL - 1);   // clamp so loads stay in range

    float tc[LL];                                    // transition column T[:, jj]
#pragma unroll
    for (int i = 0; i < LL; ++i) tc[i] = T[i * LL + jj];

    float alpha = lg[jj];                            // alpha0 = logits[b, 0, :]

    for (int t = 1; t < SS; ++t) {
        const float e = lg[t * LL + jj];
        float sc[LL];
        float m = -3.402823466e38f;
#pragma unroll
        for (int i = 0; i < LL; ++i) {
            const float ai = __shfl(alpha, i);       // broadcast old alpha[i]
            sc[i] = ai + tc[i];
            m = fmaxf(m, sc[i]);
        }
        float sum = 0.0f;
#pragma unroll
        for (int i = 0; i < LL; ++i) sum += expf(sc[i] - m);
        const float nv = m + logf(sum) + e;
        alpha = (t < seqlen) ? nv : alpha;           // wave-uniform predicate
    }

    // ---- denominator = logsumexp(alpha) over the 9 states ----
    float v = (lane < LL) ? alpha : -3.402823466e38f;
    float m = v;
#pragma unroll
    for (int off = 16; off > 0; off >>= 1) m = fmaxf(m, __shfl_xor(m, off));
    float se = expf(v - m);
#pragma unroll
    for (int off = 16; off > 0; off >>= 1) se += __shfl_xor(se, off);
    const float den = m + logf(se);

    if (lane == 0) atomicAdd(out, den - num);        // sum over batch of -(num-den)
}

// ---------------------------------------------------------------------------
extern "C" void kernel_launch(void* const* d_in, const int* in_sizes, int n_in,
                              void* d_out, int out_size, void* d_ws, size_t ws_size,
                              hipStream_t stream) {
    const float* inputs   = (const float*)d_in[0];   // (B,S,D) f32
    const int*   targets  = (const int*)  d_in[1];   // (B,S)   i32
    const int*   seqlens  = (const int*)  d_in[2];   // (B,)    i32
    // d_in[3] = mask (bool) -- recomputed from seqlens instead
    const float* W        = (const float*)d_in[4];   // (D,L)   f32
    const float* bias     = (const float*)d_in[5];   // (L,)    f32
    const float* trans    = (const float*)d_in[6];   // (L,L)   f32
    float* out = (float*)d_out;                      // [0]=nll, [1..]=logits

    crf_gemm_logits_kernel<<<dim3(256), dim3(256), 0, stream>>>(inputs, W, bias, out);
    crf_forward_kernel<<<dim3(BB), dim3(32), 0, stream>>>(targets, seqlens, trans, out);
}